// OctonionDenseLinear_12249246728417
// MI455X (gfx1250) — compile-verified
//
#include <hip/hip_runtime.h>
#include <hip/hip_bf16.h>

// ---------------------------------------------------------------------------
// Octonion structure constants, computed at compile time (mirrors reference).
// ---------------------------------------------------------------------------
struct CTab { float c[8][8][8]; };

constexpr CTab makeC() {
    CTab t{};
    t.c[0][0][0] = 1.0f;
    for (int i = 1; i < 8; ++i) {
        t.c[0][i][i] = 1.0f;
        t.c[i][0][i] = 1.0f;
        t.c[i][i][0] = -1.0f;
    }
    constexpr int tr[7][3] = {{1,2,3},{1,4,5},{1,7,6},{2,4,6},{2,5,7},{3,4,7},{3,6,5}};
    for (int n = 0; n < 7; ++n) {
        const int a = tr[n][0], b = tr[n][1], c = tr[n][2];
        const int perms[3][3] = {{a,b,c},{b,c,a},{c,a,b}};
        for (int p = 0; p < 3; ++p) {
            t.c[perms[p][0]][perms[p][1]][perms[p][2]] =  1.0f;
            t.c[perms[p][1]][perms[p][0]][perms[p][2]] = -1.0f;
        }
    }
    return t;
}

__device__ __constant__ CTab c_C = makeC();

// Problem constants (from reference)
#define M_TOTAL 65536   // 16*4096 batch rows
#define KDIM    512     // 64 in-features * 8 components
#define NDIM    512     // 64 out-features * 8 components

// GEMM tiling
#define BM 64
#define BN 128
#define BK 32
#define NCHUNK (KDIM / BK)      // 16
#define LDX (BK + 4)            // LDS row pad: 36 floats = 144B (16B aligned)
#define LDW (BK + 4)

typedef __attribute__((ext_vector_type(2))) float v2f;
typedef __attribute__((ext_vector_type(8))) float v8f;

// ---------------------------------------------------------------------------
// Kernel 1: fold structure constants into block weight.
//   Wb[n][k] = sum_i C[i][j][kc] * W[i][o][f],  n = o*8+kc, k = f*8+j
// ---------------------------------------------------------------------------
__global__ __launch_bounds__(256)
void build_wbig_kernel(const float* __restrict__ W, float* __restrict__ Wb) {
    const int idx = blockIdx.x * blockDim.x + threadIdx.x;   // 0 .. 512*512-1
    const int n  = idx >> 9;
    const int k  = idx & (KDIM - 1);
    const int o  = n >> 3, kc = n & 7;
    const int f  = k >> 3, j  = k & 7;
    float acc = 0.0f;
#pragma unroll
    for (int i = 0; i < 8; ++i)
        acc += c_C.c[i][j][kc] * W[(i * 64 + o) * 64 + f];
    Wb[idx] = acc;
}

// ---------------------------------------------------------------------------
// Kernel 2: out[m][n] = sum_k x[m][k] * Wb[n][k] + bias[n]
// 256 threads = 8 waves, BM x BN tile; K marched in BK chunks, double-buffered
// LDS filled by GLOBAL_LOAD_ASYNC_TO_LDS_B128 (ASYNCcnt), WMMA f32 16x16x4.
// Fragment layouts (ISA 7.12.2):
//   A:   lane = (half<<4)|m, vgpr v -> [m][2*half + v]
//   B:   lane = (half<<4)|n, vgpr v -> [2*half + v][n]    (K x N)
//   C/D: lane = (half<<4)|n, vgpr v -> [v + 8*half][n]
// Wb stored [n][k] (B^T) so A and B fragments load identically.
// ---------------------------------------------------------------------------
__global__ __launch_bounds__(256)
void octo_gemm_kernel(const float* __restrict__ X,
                      const float* __restrict__ Wb,
                      const float* __restrict__ bias,
                      float* __restrict__ out) {
    __shared__ float xs[2][BM][LDX];   // 2 * 64 * 36 * 4  = 18.0 KB
    __shared__ float ws[2][BN][LDW];   // 2 * 128 * 36 * 4 = 36.0 KB

    const int tid  = threadIdx.x;
    const int lane = tid & 31;
    const int wv   = tid >> 5;      // wave 0..7
    const int wm   = wv & 3;        // M subtile (16 rows each)
    const int wn   = wv >> 2;       // N group (64 cols each)
    const int half = lane >> 4;     // lane half selects K / M offset
    const int ln15 = lane & 15;     // M index for A, N index for B/C/D

    const int mBlock = blockIdx.x * BM;
    const int nBlock = blockIdx.y * BN;

    v8f acc[4] = {};

    // Issue async global->LDS copies for K-chunk k0 into LDS buffer `buf`.
    // xs tile: 64x32 floats = 512 float4 -> 2 per thread
    // ws tile: 128x32 floats = 1024 float4 -> 4 per thread
    auto issue_loads = [&](int k0, int buf) {
#pragma unroll
        for (int r = 0; r < 2; ++r) {
            const int e    = tid + r * 256;        // 0..511
            const int row  = e >> 3;               // 8 float4 per row
            const int col4 = e & 7;
            const unsigned ldsb = (unsigned)(uintptr_t)&xs[buf][row][col4 * 4];
            const unsigned goff =
                (unsigned)(((mBlock + row) * KDIM + k0 + col4 * 4) * sizeof(float));
            asm volatile("global_load_async_to_lds_b128 %0, %1, %2"
                         :: "v"(ldsb), "v"(goff), "s"(X)
                         : "memory");
        }
#pragma unroll
        for (int r = 0; r < 4; ++r) {
            const int e    = tid + r * 256;        // 0..1023
            const int row  = e >> 3;
            const int col4 = e & 7;
            const unsigned ldsb = (unsigned)(uintptr_t)&ws[buf][row][col4 * 4];
            const unsigned goff =
                (unsigned)(((nBlock + row) * KDIM + k0 + col4 * 4) * sizeof(float));
            asm volatile("global_load_async_to_lds_b128 %0, %1, %2"
                         :: "v"(ldsb), "v"(goff), "s"(Wb)
                         : "memory");
        }
    };

    // Prologue: fill buffer 0.
    issue_loads(0, 0);
    asm volatile("s_wait_asynccnt 0x0" ::: "memory");
    __syncthreads();

    for (int c = 0; c < NCHUNK; ++c) {
        const int cur = c & 1;
        // Prefetch next chunk into the alternate buffer (in flight under WMMA).
        if (c + 1 < NCHUNK) issue_loads((c + 1) * BK, cur ^ 1);

        // ---- Compute: 8 K-steps of 4, 4 N-subtiles each ----
#pragma unroll
        for (int kk = 0; kk < BK; kk += 4) {
            v2f a;
            a.x = xs[cur][wm * 16 + ln15][kk + 2 * half + 0];
            a.y = xs[cur][wm * 16 + ln15][kk + 2 * half + 1];
#pragma unroll
            for (int t = 0; t < 4; ++t) {
                v2f b;
                b.x = ws[cur][wn * 64 + t * 16 + ln15][kk + 2 * half + 0];
                b.y = ws[cur][wn * 64 + t * 16 + ln15][kk + 2 * half + 1];
                acc[t] = __builtin_amdgcn_wmma_f32_16x16x4_f32(
                    /*neg_a=*/false, a, /*neg_b=*/false, b,
                    /*c_mod=*/(short)0, acc[t],
                    /*reuse_a=*/false, /*reuse_b=*/false);
            }
        }

        // Next chunk's async copies must have landed before anyone reads them.
        asm volatile("s_wait_asynccnt 0x0" ::: "memory");
        __syncthreads();
    }

    // ---- Epilogue: bias add + store ----
#pragma unroll
    for (int t = 0; t < 4; ++t) {
        const int ncol = nBlock + wn * 64 + t * 16 + ln15;
        const float bv = bias[ncol];
#pragma unroll
        for (int v = 0; v < 8; ++v) {
            const int row = mBlock + wm * 16 + v + 8 * half;
            out[(size_t)row * NDIM + ncol] = acc[t][v] + bv;
        }
    }
}

extern "C" void kernel_launch(void* const* d_in, const int* in_sizes, int n_in,
                              void* d_out, int out_size, void* d_ws, size_t ws_size,
                              hipStream_t stream) {
    const float* x    = (const float*)d_in[0];   // [16,4096,64,8]
    const float* W    = (const float*)d_in[1];   // [8,64,64]
    const float* bias = (const float*)d_in[2];   // [64,8]
    float*       out  = (float*)d_out;           // [16,4096,64,8]
    float*       Wb   = (float*)d_ws;            // [512,512] scratch, 1 MB

    // 1) Fold structure constants into the block weight.
    build_wbig_kernel<<<dim3((NDIM * KDIM) / 256), dim3(256), 0, stream>>>(W, Wb);

    // 2) Fused GEMM + bias (async-LDS double-buffered WMMA).
    octo_gemm_kernel<<<dim3(M_TOTAL / BM, NDIM / BN), dim3(256), 0, stream>>>(x, Wb, bias, out);
}